// SelfAttention_28750511079525
// MI455X (gfx1250) — compile-verified
//
#include <hip/hip_runtime.h>
#include <hip/hip_bf16.h>
#include <cstddef>

#define B_ 32
#define T_ 2048
#define D_ 512

typedef __attribute__((ext_vector_type(2))) float v2f;
typedef __attribute__((ext_vector_type(8))) float v8f;

// ---------------------------------------------------------------------------
// Pass 1 / Pass 4: out16[b, d0:d0+16] = sum_t coeff[b,t] * x[b,t,d0:d0+16]
// via V_WMMA_F32_16X16X4_F32.  A = coeff broadcast over M (16x4),
// B = natural 4(t) x 16(d) tile of x, D rows all identical -> row 0 is result.
// grid = (B_, D_/16), block = 256 (8 waves); wave w covers t in [w*256, w*256+256).
// ---------------------------------------------------------------------------
template <bool WEIGHTED>
__global__ __launch_bounds__(256) void pool_wmma_kernel(
    const float* __restrict__ x, const float* __restrict__ coeff,
    float* __restrict__ out) {
  const int b      = blockIdx.x;          // batch
  const int d0     = blockIdx.y * 16;     // 16-wide d chunk
  const int lane   = threadIdx.x & 31;
  const int wave   = threadIdx.x >> 5;    // 0..7
  const int col    = lane & 15;           // N index (d)
  const int kh     = (lane >> 4) * 2;     // K rows held by this half-wave: {0,1} or {2,3}

  const float* xb = x + (size_t)b * T_ * D_;
  const float* cb = WEIGHTED ? (coeff + (size_t)b * T_) : nullptr;

  const int t_begin = wave * (T_ / 8);
  const int t_end   = t_begin + (T_ / 8);

  v8f c0 = {};
  v8f c1 = {};

  for (int t0 = t_begin; t0 < t_end; t0 += 8) {
    // ---- chunk 0: t0 .. t0+3 ----
    {
      v2f a;
      if (WEIGHTED) { a.x = cb[t0 + kh]; a.y = cb[t0 + kh + 1]; }
      else          { a.x = 1.0f;        a.y = 1.0f; }
      const float* xp = xb + (size_t)(t0 + kh) * D_ + d0 + col;
      v2f bm;
      bm.x = xp[0];
      bm.y = xp[D_];
      __builtin_prefetch(xp + 8 * D_, 0, 1);   // global_prefetch_b8 (speculative)
      c0 = __builtin_amdgcn_wmma_f32_16x16x4_f32(false, a, false, bm,
                                                 (short)0, c0, false, false);
    }
    // ---- chunk 1: t0+4 .. t0+7 ----
    {
      v2f a;
      if (WEIGHTED) { a.x = cb[t0 + 4 + kh]; a.y = cb[t0 + 4 + kh + 1]; }
      else          { a.x = 1.0f;            a.y = 1.0f; }
      const float* xp = xb + (size_t)(t0 + 4 + kh) * D_ + d0 + col;
      v2f bm;
      bm.x = xp[0];
      bm.y = xp[D_];
      c1 = __builtin_amdgcn_wmma_f32_16x16x4_f32(false, a, false, bm,
                                                 (short)0, c1, false, false);
    }
  }

  // Row M=0 of D lives in accumulator VGPR0, lanes 0..15 (N = lane).
  __shared__ float red[8][16];
  if (lane < 16) red[wave][lane] = c0[0] + c1[0];
  __syncthreads();
  if (threadIdx.x < 16) {
    float acc = 0.f;
#pragma unroll
    for (int wv = 0; wv < 8; ++wv) acc += red[wv][threadIdx.x];
    out[(size_t)b * D_ + d0 + threadIdx.x] = acc;
  }
}

// ---------------------------------------------------------------------------
// Pass 2: w[b,t] = <x_bt, s_b> - <x_bt, x_bt>.  One wave per row, float4
// coalesced loads, shfl_xor wave reduction.  grid = B_*T_/8, block = 256.
// ---------------------------------------------------------------------------
__global__ __launch_bounds__(256) void dots_kernel(
    const float* __restrict__ x, const float* __restrict__ s,
    float* __restrict__ w) {
  const int lane = threadIdx.x & 31;
  const int wave = threadIdx.x >> 5;
  const int r    = blockIdx.x * 8 + wave;     // global row over [B_*T_)
  const int b    = r >> 11;                    // T_ = 2048

  const float* xr = x + (size_t)r * D_;
  const float* sr = s + (size_t)b * D_;

  float dot = 0.f, sq = 0.f;
#pragma unroll
  for (int j = 0; j < 4; ++j) {
    const int col = j * 128 + lane * 4;
    const float4 xv = *(const float4*)(xr + col);
    const float4 sv = *(const float4*)(sr + col);
    dot = fmaf(xv.x, sv.x, dot); dot = fmaf(xv.y, sv.y, dot);
    dot = fmaf(xv.z, sv.z, dot); dot = fmaf(xv.w, sv.w, dot);
    sq  = fmaf(xv.x, xv.x, sq);  sq  = fmaf(xv.y, xv.y, sq);
    sq  = fmaf(xv.z, xv.z, sq);  sq  = fmaf(xv.w, xv.w, sq);
  }
#pragma unroll
  for (int off = 16; off > 0; off >>= 1) {
    dot += __shfl_xor(dot, off, 32);
    sq  += __shfl_xor(sq,  off, 32);
  }
  if (lane == 0) w[r] = dot - sq;
}

// ---------------------------------------------------------------------------
// Pass 3: softmax over t for each b.  grid = B_, block = 256 (8 vals/thread).
// ---------------------------------------------------------------------------
__global__ __launch_bounds__(256) void softmax_kernel(
    const float* __restrict__ w, float* __restrict__ p) {
  const int b   = blockIdx.x;
  const int tid = threadIdx.x;
  const float* wb = w + (size_t)b * T_;
  float* pb       = p + (size_t)b * T_;

  __shared__ float red[256];
  float vals[8];
  float lmax = -3.4e38f;
#pragma unroll
  for (int j = 0; j < 8; ++j) {
    vals[j] = wb[j * 256 + tid];
    lmax = fmaxf(lmax, vals[j]);
  }
  red[tid] = lmax;
  __syncthreads();
#pragma unroll
  for (int sft = 128; sft > 0; sft >>= 1) {
    if (tid < sft) red[tid] = fmaxf(red[tid], red[tid + sft]);
    __syncthreads();
  }
  const float bmax = red[0];
  __syncthreads();

  float lsum = 0.f;
#pragma unroll
  for (int j = 0; j < 8; ++j) {
    vals[j] = __expf(vals[j] - bmax);
    lsum += vals[j];
  }
  red[tid] = lsum;
  __syncthreads();
#pragma unroll
  for (int sft = 128; sft > 0; sft >>= 1) {
    if (tid < sft) red[tid] += red[tid + sft];
    __syncthreads();
  }
  const float inv = 1.0f / red[0];
#pragma unroll
  for (int j = 0; j < 8; ++j) pb[j * 256 + tid] = vals[j] * inv;
}

// ---------------------------------------------------------------------------
extern "C" void kernel_launch(void* const* d_in, const int* in_sizes, int n_in,
                              void* d_out, int out_size, void* d_ws, size_t ws_size,
                              hipStream_t stream) {
  (void)in_sizes; (void)n_in; (void)out_size; (void)ws_size;
  const float* x = (const float*)d_in[0];
  float* out = (float*)d_out;

  float* ws = (float*)d_ws;
  float* s_ws  = ws;                        // B_*D_   = 16384 floats
  float* w_ws  = s_ws + (size_t)B_ * D_;    // B_*T_   = 65536 floats
  float* p_ws  = w_ws + (size_t)B_ * T_;    // B_*T_   = 65536 floats

  // Pass 1: s[b,:] = sum_t x[b,t,:]  (WMMA, all-ones coefficients)
  pool_wmma_kernel<false><<<dim3(B_, D_ / 16), 256, 0, stream>>>(x, nullptr, s_ws);
  // Pass 2: w[b,t] = x.s - x.x
  dots_kernel<<<(B_ * T_) / 8, 256, 0, stream>>>(x, s_ws, w_ws);
  // Pass 3: softmax over t
  softmax_kernel<<<B_, 256, 0, stream>>>(w_ws, p_ws);
  // Pass 4: out[b,:] = sum_t p[b,t] * x[b,t,:]  (WMMA, softmax coefficients)
  pool_wmma_kernel<true><<<dim3(B_, D_ / 16), 256, 0, stream>>>(x, p_ws, out);
}